// Attention_55147380081216
// MI455X (gfx1250) — compile-verified
//
#include <hip/hip_runtime.h>
#include <math.h>

typedef __attribute__((ext_vector_type(2))) float v2f;
typedef __attribute__((ext_vector_type(8))) float v8f;

#define N_NODES  50000
#define N_EDGES  800000
#define D_FEAT   128
#define OUT_COLS 256
#define EPSF     1e-12f
#define NEG_MAXF (-3.402823466e+38f)

// ---- workspace layout (float offsets) ----
#define WS_CU    0
#define WS_CV    128
#define WS_CONST 256              // [0]=b_coef.u, [1]=b_coef.v
#define WS_AD    1024
#define WS_AS    (WS_AD + 50048)
#define WS_EMAX  (WS_AS + 50048)  // stored as monotonic-encoded uint
#define WS_DENOM (WS_EMAX + 50048)
#define WS_EBUF  (WS_DENOM + 50048)
// total ~= 1,001,216 floats (~4 MB)

// monotonic float<->uint encoding so native u32 atomicMax == float max
__device__ __forceinline__ unsigned fenc(float f) {
  int i = __float_as_int(f);
  return (unsigned)(i ^ ((i >> 31) | 0x80000000));
}
__device__ __forceinline__ float fdec(unsigned u) {
  unsigned mask = (u & 0x80000000u) ? 0x80000000u : 0xFFFFFFFFu;
  return __uint_as_float(u ^ mask);
}

// ---- kernel 0: cu = W_coef @ u, cv = W_coef @ v, bias consts ----
__global__ void prep_kernel(const float* __restrict__ Wc, const float* __restrict__ bc,
                            const float* __restrict__ Wr, float* __restrict__ ws) {
  int d = threadIdx.x; // 0..127
  float su = 0.f, sv = 0.f;
  for (int c = 0; c < D_FEAT; ++c) {
    float w = Wc[d * D_FEAT + c];
    su += w * Wr[c];
    sv += w * Wr[D_FEAT + c];
  }
  ws[WS_CU + d] = su;
  ws[WS_CV + d] = sv;
  if (d == 0) {
    float bu = 0.f, bv = 0.f;
    for (int c = 0; c < D_FEAT; ++c) { bu += bc[c] * Wr[c]; bv += bc[c] * Wr[D_FEAT + c]; }
    ws[WS_CONST]     = bu;
    ws[WS_CONST + 1] = bv;
  }
}

// ---- kernel 1: per-node scalars a_d = h.cu + const, a_s = h.cv + const (1 wave / node) ----
__global__ void node_score_kernel(const float* __restrict__ h, float* __restrict__ ws) {
  int wave = threadIdx.x >> 5;
  int lane = threadIdx.x & 31;
  int node = blockIdx.x * 8 + wave;
  if (node >= N_NODES) return;
  float4 hv = ((const float4*)(h + node * D_FEAT))[lane];
  float4 u  = ((const float4*)(ws + WS_CU))[lane];
  float4 v  = ((const float4*)(ws + WS_CV))[lane];
  float du = hv.x * u.x + hv.y * u.y + hv.z * u.z + hv.w * u.w;
  float dv = hv.x * v.x + hv.y * v.y + hv.z * v.z + hv.w * v.w;
  for (int off = 16; off > 0; off >>= 1) {
    du += __shfl_down(du, off, 32);
    dv += __shfl_down(dv, off, 32);
  }
  if (lane == 0) {
    ws[WS_AD + node] = du + ws[WS_CONST];
    ws[WS_AS + node] = dv + ws[WS_CONST + 1];
  }
}

// ---- kernel 2: init emax/denom and zero agg region (right half of d_out) ----
__global__ void init_kernel(float* __restrict__ ws, float* __restrict__ out) {
  int i = blockIdx.x * blockDim.x + threadIdx.x;
  if (i < N_NODES) {
    ((unsigned*)ws)[WS_EMAX + i] = fenc(NEG_MAXF);
    ws[WS_DENOM + i] = 0.f;
  }
  if (i < N_NODES * D_FEAT) {
    int row = i >> 7, c = i & 127;
    out[row * OUT_COLS + 128 + c] = 0.f;
  }
}

// ---- kernel 3: per-edge score + segment max ----
__global__ void edge_max_kernel(const int* __restrict__ src, const int* __restrict__ dst,
                                const float* __restrict__ brd, float* __restrict__ ws) {
  int e = blockIdx.x * blockDim.x + threadIdx.x;
  if (e >= N_EDGES) return;
  int d = dst[e];
  float ev = ws[WS_AD + d] + ws[WS_AS + src[e]] + brd[0];
  ws[WS_EBUF + e] = ev;
  atomicMax((unsigned*)ws + WS_EMAX + d, fenc(ev));
}

// ---- kernel 4: exp + segment sum ----
__global__ void edge_exp_kernel(const int* __restrict__ dst, float* __restrict__ ws) {
  int e = blockIdx.x * blockDim.x + threadIdx.x;
  if (e >= N_EDGES) return;
  int d = dst[e];
  float em = fdec(((const unsigned*)ws)[WS_EMAX + d]);
  float ex = expf(ws[WS_EBUF + e] - em);
  ws[WS_EBUF + e] = ex;
  atomicAdd(&ws[WS_DENOM + d], ex);
}

// ---- kernel 5: agg[dst] += alpha * h[src]  (1 wave / edge, agg lives in out[:,128:256]) ----
__global__ void edge_agg_kernel(const int* __restrict__ src, const int* __restrict__ dst,
                                const float* __restrict__ h, const float* __restrict__ ws,
                                float* __restrict__ out) {
  int wave = threadIdx.x >> 5;
  int lane = threadIdx.x & 31;
  int e = blockIdx.x * 8 + wave;
  if (e >= N_EDGES) return;
  int s = src[e], d = dst[e];
  float alpha = ws[WS_EBUF + e] / (ws[WS_DENOM + d] + EPSF);
  float4 x = ((const float4*)(h + s * D_FEAT))[lane];
  float* ag = out + (size_t)d * OUT_COLS + 128 + lane * 4;
  atomicAdd(ag + 0, alpha * x.x);
  atomicAdd(ag + 1, alpha * x.y);
  atomicAdd(ag + 2, alpha * x.z);
  atomicAdd(ag + 3, alpha * x.w);
}

// ---- kernel 6: fused [h@W_node+b | agg@W_neigh+b] GEMM + row L2-normalize (WMMA f32 16x16x4) ----
// 16 output rows per block; 8 waves, each wave owns two 16-col N-tiles of one half.
__global__ __launch_bounds__(256)
void gemm_norm_kernel(const float* __restrict__ h,
                      const float* __restrict__ Wnode,  const float* __restrict__ bnode,
                      const float* __restrict__ Wneigh, const float* __restrict__ bneigh,
                      float* __restrict__ out) {
  __shared__ float As[2][16 * 132];  // [0]=h tile, [1]=agg tile (padded stride 132)
  __shared__ float St[16 * 260];     // 16 x 256 output staging (padded stride 260)
  __shared__ float Sc[16];           // per-row rsqrt scales

  const int tid   = threadIdx.x;
  const int rbase = blockIdx.x * 16;

  // stage A tiles (coalesced float4): h rows and agg rows (from out right half)
  for (int v = tid; v < 512; v += 256) {
    int row = v >> 5;
    int c4  = v & 31;
    float4 hv = *(const float4*)(h + (size_t)(rbase + row) * D_FEAT + c4 * 4);
    *(float4*)(&As[0][row * 132 + c4 * 4]) = hv;
    float4 av = *(const float4*)(out + (size_t)(rbase + row) * OUT_COLS + 128 + c4 * 4);
    *(float4*)(&As[1][row * 132 + c4 * 4]) = av;
  }
  __syncthreads();

  const int wave = tid >> 5;
  const int lane = tid & 31;
  const int half = wave >> 2;        // 0: h@W_node (cols 0..127), 1: agg@W_neigh (cols 128..255)
  const int n0   = (wave & 3) * 32;  // two adjacent 16-col tiles
  const int n1   = n0 + 16;
  const int lo   = lane & 15;
  const int hi   = lane >> 4;

  const float* __restrict__ B    = half ? Wneigh : Wnode;
  const float* __restrict__ bias = half ? bneigh : bnode;
  const float* Arow = &As[half][lo * 132];

  v8f acc0 = {};
  v8f acc1 = {};
  for (int k = 0; k < D_FEAT; k += 4) {
    int ka = k + hi * 2;
    // A frag (16x4 f32): lane holds A[lo][ka], A[lo][ka+1]
    v2f a = *(const v2f*)(Arow + ka);
    // B frag (4x16 f32): lane holds B[ka][n+lo], B[ka+1][n+lo]
    const float* bp = B + (size_t)ka * D_FEAT;
    v2f b0, b1;
    b0.x = bp[n0 + lo];  b0.y = bp[D_FEAT + n0 + lo];
    b1.x = bp[n1 + lo];  b1.y = bp[D_FEAT + n1 + lo];
    acc0 = __builtin_amdgcn_wmma_f32_16x16x4_f32(false, a, false, b0, (short)0, acc0, false, false);
    acc1 = __builtin_amdgcn_wmma_f32_16x16x4_f32(false, a, false, b1, (short)0, acc1, false, false);
  }

  float bv0 = bias[n0 + lo];
  float bv1 = bias[n1 + lo];
  const int colbase = half * 128;
#pragma unroll
  for (int j = 0; j < 8; ++j) {
    int row = j + 8 * hi; // C/D layout: vgpr j, lanes 0-15 -> M=j; lanes 16-31 -> M=j+8
    St[row * 260 + colbase + n0 + lo] = acc0[j] + bv0;
    St[row * 260 + colbase + n1 + lo] = acc1[j] + bv1;
  }
  __syncthreads();

  if (tid < 16) {
    float ssum = 0.f;
    for (int c = 0; c < 256; ++c) { float x = St[tid * 260 + c]; ssum += x * x; }
    Sc[tid] = rsqrtf(fmaxf(ssum, EPSF));
  }
  __syncthreads();

  for (int v = tid; v < 1024; v += 256) {  // 16 rows x 64 float4
    int row = v >> 6, c4 = v & 63;
    float4 x = *(const float4*)(&St[row * 260 + c4 * 4]);
    float s = Sc[row];
    x.x *= s; x.y *= s; x.z *= s; x.w *= s;
    *(float4*)(out + (size_t)(rbase + row) * OUT_COLS + c4 * 4) = x;
  }
}

extern "C" void kernel_launch(void* const* d_in, const int* in_sizes, int n_in,
                              void* d_out, int out_size, void* d_ws, size_t ws_size,
                              hipStream_t stream) {
  const float* h       = (const float*)d_in[0];
  const int*   src     = (const int*)  d_in[1];
  const int*   dst     = (const int*)  d_in[2];
  const float* W_coef  = (const float*)d_in[3];
  const float* b_coef  = (const float*)d_in[4];
  const float* W_red   = (const float*)d_in[5];
  const float* b_red   = (const float*)d_in[6];
  const float* W_node  = (const float*)d_in[7];
  const float* b_node  = (const float*)d_in[8];
  const float* W_neigh = (const float*)d_in[9];
  const float* b_neigh = (const float*)d_in[10];
  float* out = (float*)d_out;
  float* ws  = (float*)d_ws;

  prep_kernel<<<1, 128, 0, stream>>>(W_coef, b_coef, W_red, ws);
  node_score_kernel<<<(N_NODES + 7) / 8, 256, 0, stream>>>(h, ws);
  init_kernel<<<(N_NODES * D_FEAT + 255) / 256, 256, 0, stream>>>(ws, out);
  edge_max_kernel<<<(N_EDGES + 255) / 256, 256, 0, stream>>>(src, dst, b_red, ws);
  edge_exp_kernel<<<(N_EDGES + 255) / 256, 256, 0, stream>>>(dst, ws);
  edge_agg_kernel<<<(N_EDGES + 7) / 8, 256, 0, stream>>>(src, dst, h, ws, out);
  gemm_norm_kernel<<<(N_NODES + 15) / 16, 256, 0, stream>>>(h, W_node, b_node, W_neigh, b_neigh, out);
}